// tfft_53214644797612
// MI455X (gfx1250) — compile-verified
//
#include <hip/hip_runtime.h>
#include <cstdint>
#include <cstddef>

// ---------------------------------------------------------------------------
// Problem constants (from reference)
// ---------------------------------------------------------------------------
constexpr int NFFT = 1024;      // FFT length
constexpr int ROWS = 8 * 512;   // BATCH * NUMF = 4096 rows
constexpr int WPB  = 4;         // waves (rows) per block
constexpr int TPB  = WPB * 32;  // 128 threads per block
// Packed weight LDS: entry e in [0,2048), float index = 4*e + 4*(e>>5) (skew
// breaks the 512B periodicity of big-stage accesses across banks).
constexpr int SWSZ = 4 * 2048 + 4 * 64;   // 8448 floats = 33 KB
// Row scratch: 1024 floats for staging/gather, 33*32 for padded transposes.
constexpr int SXSZ = 33 * 32;             // 1056 floats

// ---------------------------------------------------------------------------
// gfx1250 async global->LDS path (signature verified by compiler in R1/R2).
// ---------------------------------------------------------------------------
typedef int v4i __attribute__((vector_size(16)));
typedef __attribute__((address_space(1))) v4i* g_v4i_ptr;
typedef __attribute__((address_space(3))) v4i* l_v4i_ptr;

#if defined(__gfx1250__) && __has_builtin(__builtin_amdgcn_global_load_async_to_lds_b128)
#define USE_ASYNC 1
__device__ __forceinline__ void async_copy16(const float* g, float* l) {
  __builtin_amdgcn_global_load_async_to_lds_b128(
      (g_v4i_ptr)(void*)const_cast<float*>(g), (l_v4i_ptr)(void*)l,
      /*offset=*/0, /*cpol=*/0);
}
#else
#define USE_ASYNC 0
#endif

__device__ __forceinline__ constexpr int rev5(int v) {
  return ((v & 1) << 4) | ((v & 2) << 2) | (v & 4) | ((v & 8) >> 2) |
         ((v & 16) >> 4);
}

// ---------------------------------------------------------------------------
// Butterfly core: out0 = w0.{x,y}*(v0,v1) - w0.{z,w}*(q0,q1) etc.
// Weights packed in LDS as float4 {rw0[j], rw1[j], iw0[j], iw1[j]} at entry
// e = off(stage) + j, float index 4*e + 4*(e>>5). off(0)=0, off(i)=2^(i+1).
// ---------------------------------------------------------------------------
__device__ __forceinline__ void bfly(float& R0, float& Q0, float& R1,
                                     float& Q1, const float4 w0,
                                     const float4 w1) {
  const float r0 = R0, r1 = R1, q0 = Q0, q1 = Q1;
  R0 = w0.x * r0 + w0.y * r1 - w0.z * q0 - w0.w * q1;
  Q0 = w0.x * q0 + w0.y * q1 + w0.z * r0 + w0.w * r1;
  R1 = w1.x * r0 + w1.y * r1 - w1.z * q0 - w1.w * q1;
  Q1 = w1.x * q0 + w1.y * q1 + w1.z * r0 + w1.w * r1;
}

// Stages 0..4, contiguous layout: lane L holds n = 32*L + c.  h = 1..16,
// pairs live inside the lane's 32 registers; weight index j = c % subdim
// is lane-uniform (LDS broadcast reads).
template <int I>
__device__ __forceinline__ void stage_small(float (&r)[32], float (&q)[32],
                                            const float* sW) {
  constexpr int h   = 1 << I;
  constexpr int sub = 2 * h;
  constexpr int off = (I == 0) ? 0 : sub;
#pragma unroll
  for (int b = 0; b < 32; b += sub) {
#pragma unroll
    for (int t = 0; t < h; ++t) {
      const int c0 = b + t, c1 = c0 + h;
      const int e0 = off + t, e1 = e0 + h;
      const float4 w0 = *(const float4*)(sW + 4 * e0 + 4 * (e0 >> 5));
      const float4 w1 = *(const float4*)(sW + 4 * e1 + 4 * (e1 >> 5));
      bfly(r[c0], q[c0], r[c1], q[c1], w0, w1);
    }
  }
}

// Stages 5..9, strided layout: lane L holds n = 32*c + L.  Pair distance in
// c-space is H = h/32; weight index j = 32*t + lane (lane-consecutive float4
// reads; only 2*H unique loads per stage after CSE).
template <int I>
__device__ __forceinline__ void stage_strided(float (&r)[32], float (&q)[32],
                                              const float* sW, int lane) {
  constexpr int H   = 1 << (I - 5);
  constexpr int SUB = 2 * H;
  constexpr int off = 2 << I;           // subdim = 2^(I+1)
#pragma unroll
  for (int b = 0; b < 32; b += SUB) {
#pragma unroll
    for (int t = 0; t < H; ++t) {
      const int c0 = b + t, c1 = c0 + H;
      const int e0 = off + 32 * t + lane;
      const int e1 = e0 + 32 * H;
      const float4 w0 = *(const float4*)(sW + 4 * e0 + 4 * (e0 >> 5));
      const float4 w1 = *(const float4*)(sW + 4 * e1 + 4 * (e1 >> 5));
      bfly(r[c0], q[c0], r[c1], q[c1], w0, w1);
    }
  }
}

// Wave-private 32x32 register transpose through padded LDS (stride 33:
// gcd(33,64)=1 -> store banks 33L mod 64 all distinct; read is consecutive).
__device__ __forceinline__ void transpose_via_lds(float (&v)[32], float* sx,
                                                  int lane) {
#pragma unroll
  for (int c = 0; c < 32; ++c) sx[33 * lane + c] = v[c];
#pragma unroll
  for (int c = 0; c < 32; ++c) v[c] = sx[33 * c + lane];
}

// ---------------------------------------------------------------------------
// Kernel: one wave per row. Whole row lives in registers; single HBM pass.
// ---------------------------------------------------------------------------
__global__ void __launch_bounds__(TPB) tfft_butterfly_kernel(
    const float* __restrict__ x, const float* __restrict__ window,
    const float* __restrict__ rconvs, const float* __restrict__ iconvs,
    float* __restrict__ out) {
  __shared__ alignas(16) float sW[SWSZ];
  __shared__ alignas(16) float sWin[NFFT];
  __shared__ alignas(16) float sX[WPB][SXSZ];

  const int tid = threadIdx.x;

  // ---- Stage compacted, interleaved weights into LDS (once per block) ----
  for (int i = 0; i < 10; ++i) {
    const int sub = 2 << i;                  // subdim = 2^(i+1)
    const int off = (i == 0) ? 0 : sub;      // packed offset
    for (int j = tid; j < sub; j += TPB) {
      const int e  = off + j;
      const int fi = 4 * e + 4 * (e >> 5);
      sW[fi + 0] = rconvs[(2 * i + 0) * NFFT + j];
      sW[fi + 1] = rconvs[(2 * i + 1) * NFFT + j];
      sW[fi + 2] = iconvs[(2 * i + 0) * NFFT + j];
      sW[fi + 3] = iconvs[(2 * i + 1) * NFFT + j];
    }
  }
  for (int j = tid; j < NFFT; j += TPB) sWin[j] = window[j];
  __syncthreads();

  // ---- Per-wave row setup ----
  const int lane = tid & 31;
  const int wv   = tid >> 5;
  const int row  = blockIdx.x * WPB + wv;    // grid sized exactly: no bounds
  const float* xrow = x + (size_t)row * NFFT;
  float* sx = sX[wv];

  // Stage the row into wave-private LDS (coalesced 512B per instruction).
#if USE_ASYNC
#pragma unroll
  for (int k = 0; k < 8; ++k)
    async_copy16(xrow + 128 * k + 4 * lane, sx + 128 * k + 4 * lane);
  asm volatile("s_wait_asynccnt 0x0" ::: "memory");
#else
#pragma unroll
  for (int k = 0; k < 8; ++k)
    *(float4*)(sx + 128 * k + 4 * lane) =
        *(const float4*)(xrow + 128 * k + 4 * lane);
#endif
  __syncthreads();

  // ---- Window + bit-reverse gather (conflict-free: addr = 32*rev5(c)+rev5(L))
  float r[32], q[32];
  const int rl = rev5(lane);
#pragma unroll
  for (int c = 0; c < 32; ++c) {
    const int a = 32 * rev5(c) + rl;         // rev10(32*lane + c)
    r[c] = sx[a] * sWin[a];
    q[c] = 0.0f;
  }

  // ---- Stages 0..4 in contiguous layout ----
  stage_small<0>(r, q, sW);
  stage_small<1>(r, q, sW);
  stage_small<2>(r, q, sW);
  stage_small<3>(r, q, sW);
  stage_small<4>(r, q, sW);

  // ---- Re-partition to strided layout (n = 32*c + lane); wave-private,
  //      same-wave LDS ops are in-order so no barriers are needed. ----
  transpose_via_lds(r, sx, lane);
  transpose_via_lds(q, sx, lane);

  // ---- Stages 5..9 as intra-lane pair butterflies ----
  stage_strided<5>(r, q, sW, lane);
  stage_strided<6>(r, q, sW, lane);
  stage_strided<7>(r, q, sW, lane);
  stage_strided<8>(r, q, sW, lane);
  stage_strided<9>(r, q, sW, lane);

  // ---- Store rx then ix: strided layout => every b32 store instruction
  //      writes one contiguous, fully-coalesced 128B line. ----
  float* outr = out + (size_t)row * NFFT + lane;
  float* outi = outr + (size_t)ROWS * NFFT;
#pragma unroll
  for (int c = 0; c < 32; ++c) {
    outr[32 * c] = r[c];
    outi[32 * c] = q[c];
  }
}

// ---------------------------------------------------------------------------
// Host entry
// ---------------------------------------------------------------------------
extern "C" void kernel_launch(void* const* d_in, const int* in_sizes, int n_in,
                              void* d_out, int out_size, void* d_ws,
                              size_t ws_size, hipStream_t stream) {
  (void)in_sizes; (void)n_in; (void)out_size; (void)d_ws; (void)ws_size;
  const float* x      = (const float*)d_in[0];
  const float* window = (const float*)d_in[1];
  // d_in[2] is the 1024x1024 permutation matrix: implemented analytically
  // as the 10-bit bit-reverse involution, so it is not read.
  const float* rconvs = (const float*)d_in[3];
  const float* iconvs = (const float*)d_in[4];
  float* out = (float*)d_out;

  dim3 grid(ROWS / WPB);   // 1024 blocks
  dim3 block(TPB);         // 128 threads = 4 waves
  tfft_butterfly_kernel<<<grid, block, 0, stream>>>(x, window, rconvs, iconvs,
                                                    out);
}